// GNNModel_20409684590781
// MI455X (gfx1250) — compile-verified
//
#include <hip/hip_runtime.h>
#include <hip/hip_bf16.h>

typedef float v2f __attribute__((ext_vector_type(2)));
typedef float v8f __attribute__((ext_vector_type(8)));

static __host__ int cdiv(int a, int b) { return (a + b - 1) / b; }

__global__ void zero_f(float* __restrict__ p, int n) {
  int i = blockIdx.x * blockDim.x + threadIdx.x;
  if (i < n) p[i] = 0.0f;
}

// deg[dst] += 1 per edge (float atomics; deg accumulated in dinv buffer)
__global__ void edge_degree(const int* __restrict__ dst, int E, float* __restrict__ degf) {
  int e = blockIdx.x * blockDim.x + threadIdx.x;
  if (e < E) atomicAdd(&degf[dst[e]], 1.0f);
}

// in-place: dinv[i] = rsqrt(deg[i] + 1)   (+1 = self loop)
__global__ void make_dinv(float* __restrict__ dinv, int N) {
  int i = blockIdx.x * blockDim.x + threadIdx.x;
  if (i < N) dinv[i] = rsqrtf(dinv[i] + 1.0f);
}

// G[row, col] = dinv[row] * sum_k X[row,k] * W[k,col]
// X: [N,K] row-major, W: [K,16], G: [N,16]. One wave per 16-row tile, f32 WMMA.
// A 16x4: lane l (l<16) holds row l, k = k0+{0,1}; lane l+16 holds row l, k = k0+{2,3}.
// B 4x16: lane l holds col l&15, k = k0 + 2*(l>>4) + {0,1}.
// C 16x16: vgpr j, lanes 0-15 -> (m=j, n=lane), lanes 16-31 -> (m=j+8, n=lane-16).
// Boundary handling is value-masked (no divergent branches around loads): OOB rows
// read row 0 and multiply A by 0; B never needs masking since A==0 kills the product.
__global__ void gemm_tile16(const float* __restrict__ X, const float* __restrict__ W,
                            const float* __restrict__ dinv, float* __restrict__ G,
                            int N, int K, int tiles) {
  int tile = blockIdx.x * (blockDim.x >> 5) + (threadIdx.x >> 5);  // wave-uniform
  if (tile >= tiles) return;                                       // whole wave exits
  int lane = threadIdx.x & 31;
  int hi = lane >> 4;   // 0 or 1
  int lo = lane & 15;
  int row = tile * 16 + lo;
  float rowMask = (row < N) ? 1.0f : 0.0f;
  const float* xr = X + (size_t)((row < N) ? row : 0) * K;
  const float* wc = W + lo;  // column lo of W, row stride 16
  int K4 = K & ~3;

  v8f c = {};
  int k0 = 0;
  for (; k0 < K4; k0 += 4) {
    int kk = k0 + 2 * hi;
    if ((k0 & 63) == 0) __builtin_prefetch(xr + kk + 64, 0, 1);  // global_prefetch_b8
    v2f a, b;
    a.x = xr[kk] * rowMask;
    a.y = xr[kk + 1] * rowMask;
    b.x = wc[kk * 16];
    b.y = wc[(kk + 1) * 16];
    c = __builtin_amdgcn_wmma_f32_16x16x4_f32(false, a, false, b, (short)0, c,
                                              false, false);
  }
  if (k0 < K) {  // uniform tail (K % 4 != 0): clamp addresses, mask A values
    int kk = k0 + 2 * hi;
    int kc0 = (kk < K) ? kk : 0;
    int kc1 = (kk + 1 < K) ? kk + 1 : 0;
    float m0 = (kk < K) ? rowMask : 0.0f;
    float m1 = (kk + 1 < K) ? rowMask : 0.0f;
    v2f a, b;
    a.x = xr[kc0] * m0;
    a.y = xr[kc1] * m1;
    b.x = wc[kc0 * 16];
    b.y = wc[kc1 * 16];
    c = __builtin_amdgcn_wmma_f32_16x16x4_f32(false, a, false, b, (short)0, c,
                                              false, false);
  }
#pragma unroll
  for (int j = 0; j < 8; ++j) {
    int rr = tile * 16 + 8 * hi + j;  // m = j + 8*hi
    if (rr < N) G[(size_t)rr * 16 + lo] = c[j] * dinv[rr];
  }
}

// agg[dst] += g[src]  (16 floats per edge; 4 threads/edge, float4 gather + 4 atomics)
__global__ void edge_agg(const int* __restrict__ src, const int* __restrict__ dst,
                         const float* __restrict__ g, float* __restrict__ agg, int E4) {
  int t = blockIdx.x * blockDim.x + threadIdx.x;
  if (t >= E4) return;
  int e = t >> 2, part = t & 3;
  int s = src[e], d = dst[e];
  float4 v = ((const float4*)(g + (size_t)s * 16))[part];
  float* o = agg + (size_t)d * 16 + part * 4;
  atomicAdd(o + 0, v.x);
  atomicAdd(o + 1, v.y);
  atomicAdd(o + 2, v.z);
  atomicAdd(o + 3, v.w);
}

// h[i,:] = relu(dinv[i]*(agg[i,:] + g[i,:]) + b[:])
__global__ void post1(const float* __restrict__ agg, const float* __restrict__ g,
                      const float* __restrict__ dinv, const float* __restrict__ b,
                      float* __restrict__ h, int N) {
  int i = blockIdx.x * blockDim.x + threadIdx.x;
  if (i >= N) return;
  float di = dinv[i];
  const float4* av = (const float4*)(agg + (size_t)i * 16);
  const float4* gv = (const float4*)(g + (size_t)i * 16);
  float4* hv = (float4*)(h + (size_t)i * 16);
#pragma unroll
  for (int j = 0; j < 4; ++j) {
    float4 a = av[j], gg = gv[j];
    float4 r;
    r.x = fmaxf(di * (a.x + gg.x) + b[j * 4 + 0], 0.0f);
    r.y = fmaxf(di * (a.y + gg.y) + b[j * 4 + 1], 0.0f);
    r.z = fmaxf(di * (a.z + gg.z) + b[j * 4 + 2], 0.0f);
    r.w = fmaxf(di * (a.w + gg.w) + b[j * 4 + 3], 0.0f);
    hv[j] = r;
  }
}

// h2 = relu(dinv*(agg+g)+b2); p[i] = h2 . Wfc[0:16]; q[i] = h2 . Wfc[16:32]
__global__ void post2(const float* __restrict__ agg, const float* __restrict__ g,
                      const float* __restrict__ dinv, const float* __restrict__ b,
                      const float* __restrict__ Wfc, float* __restrict__ p,
                      float* __restrict__ q, int N) {
  int i = blockIdx.x * blockDim.x + threadIdx.x;
  if (i >= N) return;
  float di = dinv[i];
  float sp = 0.0f, sq = 0.0f;
#pragma unroll
  for (int f = 0; f < 16; ++f) {
    float h = fmaxf(di * (agg[(size_t)i * 16 + f] + g[(size_t)i * 16 + f]) + b[f], 0.0f);
    sp += h * Wfc[f];
    sq += h * Wfc[16 + f];
  }
  p[i] = sp;
  q[i] = sq;
}

// out[e] = p[src[e]] + q[dst[e]] + bfc
__global__ void edge_out(const int* __restrict__ src, const int* __restrict__ dst,
                         const float* __restrict__ p, const float* __restrict__ q,
                         const float* __restrict__ bfc, float* __restrict__ out, int E) {
  int e = blockIdx.x * blockDim.x + threadIdx.x;
  if (e < E) out[e] = p[src[e]] + q[dst[e]] + bfc[0];
}

extern "C" void kernel_launch(void* const* d_in, const int* in_sizes, int n_in,
                              void* d_out, int out_size, void* d_ws, size_t ws_size,
                              hipStream_t stream) {
  const float* x   = (const float*)d_in[0];
  const int*   ei  = (const int*)d_in[1];
  const float* W1  = (const float*)d_in[2];
  const float* b1  = (const float*)d_in[3];
  const float* W2  = (const float*)d_in[4];
  const float* b2  = (const float*)d_in[5];
  const float* Wfc = (const float*)d_in[6];
  const float* bfc = (const float*)d_in[7];
  float* out = (float*)d_out;

  const int K1 = 495;
  const int N = in_sizes[0] / K1;
  const int E = in_sizes[1] / 2;
  const int* src = ei;
  const int* dst = ei + E;

  // workspace layout (floats): dinv[N] | g[16N] | agg[16N] | h1[16N] | p[N] | q[N]
  float* ws  = (float*)d_ws;
  float* dinv = ws;
  float* g    = ws + (size_t)N;
  float* agg  = ws + (size_t)N * 17;
  float* h1   = ws + (size_t)N * 33;
  float* p    = ws + (size_t)N * 49;
  float* q    = ws + (size_t)N * 50;

  const int B = 256;
  const int tiles = cdiv(N, 16);
  const int gemmBlocks = cdiv(tiles, B / 32);

  // degrees -> dinv
  zero_f<<<cdiv(N, B), B, 0, stream>>>(dinv, N);
  edge_degree<<<cdiv(E, B), B, 0, stream>>>(dst, E, dinv);
  make_dinv<<<cdiv(N, B), B, 0, stream>>>(dinv, N);

  // layer 1: g = dinv .* (x @ W1); agg = scatter(g, src->dst); h1 = relu(dinv.*(agg+g)+b1)
  gemm_tile16<<<gemmBlocks, B, 0, stream>>>(x, W1, dinv, g, N, K1, tiles);
  zero_f<<<cdiv(16 * N, B), B, 0, stream>>>(agg, 16 * N);
  edge_agg<<<cdiv(4 * E, B), B, 0, stream>>>(src, dst, g, agg, 4 * E);
  post1<<<cdiv(N, B), B, 0, stream>>>(agg, g, dinv, b1, h1, N);

  // layer 2: g = dinv .* (h1 @ W2); agg = scatter; fused relu + per-node head dots
  gemm_tile16<<<gemmBlocks, B, 0, stream>>>(h1, W2, dinv, g, N, 16, tiles);
  zero_f<<<cdiv(16 * N, B), B, 0, stream>>>(agg, 16 * N);
  edge_agg<<<cdiv(4 * E, B), B, 0, stream>>>(src, dst, g, agg, 4 * E);
  post2<<<cdiv(N, B), B, 0, stream>>>(agg, g, dinv, b2, Wfc, p, q, N);

  // per-edge output
  edge_out<<<cdiv(E, B), B, 0, stream>>>(src, dst, p, q, bfc, out, E);
}